// GATBlock_82575041233133
// MI455X (gfx1250) — compile-verified
//
#include <hip/hip_runtime.h>
#include <hip/hip_bf16.h>

// ---------------- problem constants (match reference) ----------------
#define N_NODES 50000
#define N_EDGES 800000
#define IN_DIM  256
#define HC      128      // HEADS*OUT_DIM
#define HEADS   4
#define CH      32       // OUT_DIM per head
#define NEG_SLOPE 0.2f
#define LN_EPS  1e-5f

typedef float v2f __attribute__((ext_vector_type(2)));
typedef float v8f __attribute__((ext_vector_type(8)));

// ---------------------------------------------------------------------
// K1: h = x @ W   (50000x256 @ 256x128), f32 WMMA 16x16x4.
// One wave computes one 16x16 output tile; block = 8 waves = the 8
// column tiles of one 16-row stripe. grid.x = 50000/16 = 3125.
// ---------------------------------------------------------------------
__global__ __launch_bounds__(256)
void k_gemm(const float* __restrict__ x, const float* __restrict__ W,
            float* __restrict__ h_buf) {
    const int wave = threadIdx.x >> 5;          // 0..7 -> column tile
    const int lane = threadIdx.x & 31;
    const int row0 = blockIdx.x * 16;
    const int col0 = wave * 16;
    const int r    = lane & 15;                 // row within A tile / col within B,C,D
    const int kh   = (lane >> 4) * 2;           // K pair selector: lanes 0-15 -> K{0,1}, 16-31 -> K{2,3}

    v8f acc = {};
    #pragma unroll 4
    for (int kc = 0; kc < IN_DIM; kc += 4) {
        // A fragment: x[row0+r][kc+kh], x[row0+r][kc+kh+1]  (ISA 16x4 f32 layout)
        const float* ap = x + (size_t)(row0 + r) * IN_DIM + kc + kh;
        v2f a; a.x = ap[0]; a.y = ap[1];
        // B fragment: W[kc+kh][col0+r], W[kc+kh+1][col0+r]
        const float* bp = W + (size_t)(kc + kh) * HC + col0 + r;
        v2f b; b.x = bp[0]; b.y = bp[HC];
        acc = __builtin_amdgcn_wmma_f32_16x16x4_f32(
            /*neg_a=*/false, a, /*neg_b=*/false, b,
            /*c_mod=*/(short)0, acc, /*reuse_a=*/false, /*reuse_b=*/false);
    }
    // D layout: VGPR v -> row (v + 8*(lane>>4)), col (lane&15)
    const int mo = 8 * (lane >> 4);
    float* hp = h_buf + (size_t)(row0 + mo) * HC + col0 + (lane & 15);
    #pragma unroll
    for (int v = 0; v < 8; ++v) hp[(size_t)v * HC] = acc[v];
}

// ---------------------------------------------------------------------
// K2: per-node attention scores + self-loop logit, seed emax.
// One wave per node; lane = channel within head; wave32 shuffle reduce.
// ---------------------------------------------------------------------
__global__ __launch_bounds__(256)
void k_scores(const float* __restrict__ h_buf,
              const float* __restrict__ att_src, const float* __restrict__ att_dst,
              float* __restrict__ a_s, float* __restrict__ a_d,
              float* __restrict__ e_self, float* __restrict__ emax) {
    const int node = blockIdx.x * 8 + (threadIdx.x >> 5);
    if (node >= N_NODES) return;
    const int lane = threadIdx.x & 31;
    const float* hp = h_buf + (size_t)node * HC;
    #pragma unroll
    for (int hd = 0; hd < HEADS; ++hd) {
        float v  = hp[hd * CH + lane];
        float ps = v * att_src[hd * CH + lane];
        float pd = v * att_dst[hd * CH + lane];
        #pragma unroll
        for (int off = 16; off > 0; off >>= 1) {
            ps += __shfl_xor(ps, off, 32);
            pd += __shfl_xor(pd, off, 32);
        }
        if (lane == 0) {
            const int i = node * HEADS + hd;
            a_s[i] = ps;
            a_d[i] = pd;
            float e = ps + pd;
            e = (e > 0.0f) ? e : NEG_SLOPE * e;
            e_self[i] = e;
            emax[i]   = e;   // self-loop guarantees a valid seed (no -inf init)
        }
    }
}

// ---------------------------------------------------------------------
// float atomic max (CAS loop; values may be negative)
// ---------------------------------------------------------------------
__device__ inline void atomicMaxF(float* addr, float val) {
    int* ia = (int*)addr;
    int cur = __float_as_int(*addr);
    while (__int_as_float(cur) < val) {
        int old = atomicCAS(ia, cur, __float_as_int(val));
        if (old == cur) break;
        cur = old;
    }
}

// K3: per (edge, head): logit -> e_buf, atomic max into emax[dst]
__global__ __launch_bounds__(256)
void k_edge_max(const long long* __restrict__ ei,
                const float* __restrict__ a_s, const float* __restrict__ a_d,
                float* __restrict__ e_buf, float* __restrict__ emax) {
    const int tid = blockIdx.x * blockDim.x + threadIdx.x;
    if (tid >= N_EDGES * HEADS) return;
    const int edge = tid >> 2;
    const int hd   = tid & 3;
    const int src  = (int)ei[edge];
    const int dst  = (int)ei[N_EDGES + edge];
    float e = a_s[src * HEADS + hd] + a_d[dst * HEADS + hd];
    e = (e > 0.0f) ? e : NEG_SLOPE * e;
    e_buf[tid] = e;
    atomicMaxF(&emax[dst * HEADS + hd], e);
}

// K4: denom init with the self-loop term
__global__ __launch_bounds__(256)
void k_denom_init(const float* __restrict__ e_self, const float* __restrict__ emax,
                  float* __restrict__ denom) {
    const int i = blockIdx.x * blockDim.x + threadIdx.x;
    if (i >= N_NODES * HEADS) return;
    denom[i] = __expf(e_self[i] - emax[i]);
}

// K5: per (edge, head): ee = exp(e - emax[dst]); stash; atomicAdd denom[dst]
__global__ __launch_bounds__(256)
void k_edge_sum(const long long* __restrict__ ei,
                float* __restrict__ e_buf, const float* __restrict__ emax,
                float* __restrict__ denom) {
    const int tid = blockIdx.x * blockDim.x + threadIdx.x;
    if (tid >= N_EDGES * HEADS) return;
    const int edge = tid >> 2;
    const int hd   = tid & 3;
    const int dst  = (int)ei[N_EDGES + edge];
    float ee = __expf(e_buf[tid] - emax[dst * HEADS + hd]);
    e_buf[tid] = ee;
    atomicAdd(&denom[dst * HEADS + hd], ee);
}

// K6: out_pre[n][c] = bias[c] + alpha_self * h[n][c]
__global__ __launch_bounds__(256)
void k_out_init(const float* __restrict__ h_buf, const float* __restrict__ bias,
                const float* __restrict__ e_self, const float* __restrict__ emax,
                const float* __restrict__ denom, float* __restrict__ out_pre) {
    const int tid = blockIdx.x * blockDim.x + threadIdx.x;
    if (tid >= N_NODES * HC) return;
    const int n  = tid >> 7;
    const int c  = tid & (HC - 1);
    const int hd = c >> 5;
    const int i  = n * HEADS + hd;
    const float alpha = __expf(e_self[i] - emax[i]) / denom[i];
    out_pre[tid] = bias[c] + alpha * h_buf[tid];
}

// K7: wave per edge; lane covers 4 channels (float4 gather of h[src]),
// 4 float atomicAdds into out_pre[dst]. All buffers L2-resident.
__global__ __launch_bounds__(256)
void k_edge_aggr(const long long* __restrict__ ei,
                 const float* __restrict__ h_buf, const float* __restrict__ e_buf,
                 const float* __restrict__ denom, float* __restrict__ out_pre) {
    const int edge = blockIdx.x * 8 + (threadIdx.x >> 5);
    if (edge >= N_EDGES) return;
    const int lane = threadIdx.x & 31;
    const int src  = (int)ei[edge];
    const int dst  = (int)ei[N_EDGES + edge];
    const int hd   = lane >> 3;                  // (lane*4) >> 5
    const float alpha = e_buf[edge * HEADS + hd] / denom[dst * HEADS + hd];
    const float4 hv = *(const float4*)(h_buf + (size_t)src * HC + lane * 4);
    float* op = out_pre + (size_t)dst * HC + lane * 4;
    atomicAdd(op + 0, alpha * hv.x);
    atomicAdd(op + 1, alpha * hv.y);
    atomicAdd(op + 2, alpha * hv.z);
    atomicAdd(op + 3, alpha * hv.w);
}

// K8: LayerNorm over 128 features + ELU; wave per node -> d_out
__global__ __launch_bounds__(256)
void k_ln_elu(const float* __restrict__ out_pre,
              const float* __restrict__ gamma, const float* __restrict__ beta,
              float* __restrict__ out) {
    const int node = blockIdx.x * 8 + (threadIdx.x >> 5);
    if (node >= N_NODES) return;
    const int lane = threadIdx.x & 31;
    const float4 v = *(const float4*)(out_pre + (size_t)node * HC + lane * 4);
    float s  = v.x + v.y + v.z + v.w;
    float s2 = v.x * v.x + v.y * v.y + v.z * v.z + v.w * v.w;
    #pragma unroll
    for (int off = 16; off > 0; off >>= 1) {
        s  += __shfl_xor(s,  off, 32);
        s2 += __shfl_xor(s2, off, 32);
    }
    const float mu  = s * (1.0f / HC);
    const float var = s2 * (1.0f / HC) - mu * mu;
    const float rstd = rsqrtf(var + LN_EPS);
    float r[4] = {v.x, v.y, v.z, v.w};
    float4 o;
    float* po = &o.x;
    #pragma unroll
    for (int j = 0; j < 4; ++j) {
        const int c = lane * 4 + j;
        float y = (r[j] - mu) * rstd * gamma[c] + beta[c];
        po[j] = (y > 0.0f) ? y : (__expf(y) - 1.0f);
    }
    *(float4*)(out + (size_t)node * HC + lane * 4) = o;
}

// ---------------------------------------------------------------------
extern "C" void kernel_launch(void* const* d_in, const int* in_sizes, int n_in,
                              void* d_out, int out_size, void* d_ws, size_t ws_size,
                              hipStream_t stream) {
    const float*     x       = (const float*)d_in[0];
    const long long* ei      = (const long long*)d_in[1];   // int64 [2, E]
    const float*     W       = (const float*)d_in[2];
    const float*     att_src = (const float*)d_in[3];
    const float*     att_dst = (const float*)d_in[4];
    const float*     bias    = (const float*)d_in[5];
    const float*     gamma   = (const float*)d_in[6];
    const float*     beta    = (const float*)d_in[7];
    float* out = (float*)d_out;

    // workspace layout (floats)
    float* ws      = (float*)d_ws;
    float* h_buf   = ws;                           // N*HC            = 6,400,000
    float* a_s     = h_buf   + (size_t)N_NODES * HC;   // N*H         =   200,000
    float* a_d     = a_s     + (size_t)N_NODES * HEADS;
    float* e_self  = a_d     + (size_t)N_NODES * HEADS;
    float* emax    = e_self  + (size_t)N_NODES * HEADS;
    float* denom   = emax    + (size_t)N_NODES * HEADS;
    float* e_buf   = denom   + (size_t)N_NODES * HEADS; // E*H        = 3,200,000
    float* out_pre = e_buf   + (size_t)N_EDGES * HEADS; // N*HC       = 6,400,000

    const int B = 256;
    // K1: GEMM (3125 row tiles x 8 waves)
    k_gemm<<<N_NODES / 16, B, 0, stream>>>(x, W, h_buf);
    // K2: scores + self-loop seed (wave per node)
    k_scores<<<(N_NODES + 7) / 8, B, 0, stream>>>(h_buf, att_src, att_dst,
                                                  a_s, a_d, e_self, emax);
    // K3: edge max
    k_edge_max<<<(N_EDGES * HEADS + B - 1) / B, B, 0, stream>>>(ei, a_s, a_d, e_buf, emax);
    // K4: denom init (self-loop term)
    k_denom_init<<<(N_NODES * HEADS + B - 1) / B, B, 0, stream>>>(e_self, emax, denom);
    // K5: edge exp + denom sum
    k_edge_sum<<<(N_EDGES * HEADS + B - 1) / B, B, 0, stream>>>(ei, e_buf, emax, denom);
    // K6: out init = bias + self-loop message
    k_out_init<<<(N_NODES * HC + B - 1) / B, B, 0, stream>>>(h_buf, bias, e_self, emax,
                                                             denom, out_pre);
    // K7: edge aggregation (wave per edge)
    k_edge_aggr<<<(N_EDGES + 7) / 8, B, 0, stream>>>(ei, h_buf, e_buf, denom, out_pre);
    // K8: LayerNorm + ELU (wave per node)
    k_ln_elu<<<(N_NODES + 7) / 8, B, 0, stream>>>(out_pre, gamma, beta, out);
}